// quantiser_79216376807507
// MI455X (gfx1250) — compile-verified
//
#include <hip/hip_runtime.h>
#include <math.h>

typedef __attribute__((ext_vector_type(2))) float v2f;
typedef __attribute__((ext_vector_type(8))) float v8f;

#define N_STATES 65536
#define LATENT   512
#define KDIM     1024                 // 2*LATENT interleaved (mu,sig)
#define TILE_M   16
#define N_TILES  (N_STATES / TILE_M)  // 4096
#define MU_STRIDE 516                 // padded LDS row stride (floats) -> conflict-free
#define WSTRIDE  (LATENT + 2)         // per-wave partial: [max, sum, q[512]]

// ---------------------------------------------------------------------------
// Main pass: one wave (32 lanes) per workgroup. Each wave streams its tiles of
// 16 codebook rows once: WMMA f32 16x16x4 computes u.v dots, VALU rides along
// for ||v||^2, mus are staged to LDS for the online-softmax weighted accum.
// ---------------------------------------------------------------------------
__global__ __launch_bounds__(32)
void wq_main_kernel(const float* __restrict__ on_states,
                    const float* __restrict__ input_mu,
                    const float* __restrict__ input_sig,
                    float* __restrict__ dists_out,   // -> d_out + 514
                    float* __restrict__ ws,
                    int tiles_per_wave)
{
    __shared__ float u_s[KDIM];                    // interleaved query vector
    __shared__ float mu_s[TILE_M * MU_STRIDE];     // staged mus of current tile

    const int lane = threadIdx.x;                  // 0..31
    const int wid  = blockIdx.x;

    // stage query: u[2d] = mu[d], u[2d+1] = sig[d]
    for (int d = lane; d < LATENT; d += 32) {
        u_s[2 * d]     = input_mu[d];
        u_s[2 * d + 1] = input_sig[d];
    }
    __syncthreads();

    // ||u||^2, uniform across the wave
    float cU = 0.f;
    for (int i = lane; i < KDIM; i += 32) { float t = u_s[i]; cU += t * t; }
    #pragma unroll
    for (int off = 16; off > 0; off >>= 1) cU += __shfl_xor(cU, off, 32);

    const int half  = lane >> 4;      // 0: K pair {0,1}, 1: K pair {2,3}
    const int khalf = half << 1;
    const int mrow  = lane & 15;      // my A-matrix row within the tile

    // per-wave online softmin state (q[512] lives in VGPRs: 16 per lane)
    float q[16];
    #pragma unroll
    for (int j = 0; j < 16; ++j) q[j] = 0.f;
    float curM = -INFINITY;
    float ssum = 0.f;

    const int tile0 = wid * tiles_per_wave;
    for (int t = 0; t < tiles_per_wave; ++t) {
        const int tb = (tile0 + t) * TILE_M;
        const float* rowp = on_states + (size_t)(tb + mrow) * KDIM + khalf;

        v8f  acc = {};
        float nrm = 0.f;

        for (int cc = 0; cc < KDIM / 4; cc += 8) {
            if (cc + 8 < KDIM / 4)
                __builtin_prefetch((const void*)(rowp + 4 * (cc + 8)), 0, 3);
            #pragma unroll
            for (int c = cc; c < cc + 8; ++c) {
                // A tile: lane<16 holds row's K{0,1}, lane>=16 holds K{2,3}
                const float2 a = *(const float2*)(rowp + 4 * c);
                // B tile: query chunk broadcast to all 16 columns (same K split)
                const float2 b = *(const float2*)(&u_s[4 * c + khalf]);
                nrm += a.x * a.x + a.y * a.y;
                // a.x is always a mu element (even interleave index); stage it
                mu_s[mrow * MU_STRIDE + 2 * c + half] = a.x;
                v2f av; av.x = a.x; av.y = a.y;
                v2f bv; bv.x = b.x; bv.y = b.y;
                acc = __builtin_amdgcn_wmma_f32_16x16x4_f32(
                          false, av, false, bv, (short)0, acc, false, false);
            }
        }

        // row norms: lane l and l+16 each hold half of row (l&15)'s norm
        float n2 = nrm + __shfl_xor(nrm, 16, 32);

        // dots: B columns are identical, so lane 0 / lane 16 broadcast works
        float dist[16];
        #pragma unroll
        for (int r = 0; r < 8; ++r) {
            float dlo = __shfl(acc[r], 0, 32);    // dot(u, row r)
            float dhi = __shfl(acc[r], 16, 32);   // dot(u, row r+8)
            dist[r]     = cU - 2.f * dlo;
            dist[r + 8] = cU - 2.f * dhi;
        }
        #pragma unroll
        for (int m = 0; m < 16; ++m) dist[m] += __shfl(n2, m, 32);

        // write dists (uniform values; lanes 0..15 each pick their row)
        float dmine = dist[0];
        #pragma unroll
        for (int m = 1; m < 16; ++m) dmine = (mrow == m) ? dist[m] : dmine;
        if (lane < 16) dists_out[tb + lane] = dmine;

        // online softmin update (all values uniform across the wave)
        float tmax = -dist[0];
        #pragma unroll
        for (int m = 1; m < 16; ++m) tmax = fmaxf(tmax, -dist[m]);
        float newM  = fmaxf(curM, tmax);
        float scale = __expf(curM - newM);       // 0 on first tile (curM=-inf)
        float e[16];
        float esum = 0.f;
        #pragma unroll
        for (int m = 0; m < 16; ++m) { e[m] = __expf(-dist[m] - newM); esum += e[m]; }
        ssum = ssum * scale + esum;
        curM = newM;

        #pragma unroll
        for (int j = 0; j < 16; ++j) q[j] *= scale;
        __syncthreads();   // LDS staged mus visible (single-wave WG: cheap)
        #pragma unroll
        for (int m = 0; m < 16; ++m) {
            const float em = e[m];
            #pragma unroll
            for (int j = 0; j < 16; ++j)
                q[j] = fmaf(em, mu_s[m * MU_STRIDE + j * 32 + lane], q[j]);
        }
    }

    // emit per-wave partial [max, sum, q[512]]
    float* wb = ws + (size_t)wid * WSTRIDE;
    if (lane == 0) { wb[0] = curM; wb[1] = ssum; }
    #pragma unroll
    for (int j = 0; j < 16; ++j) wb[2 + j * 32 + lane] = q[j];
}

// ---------------------------------------------------------------------------
// Merge: log-sum-exp combine of W per-wave partials, divide, losses.
// ---------------------------------------------------------------------------
#define MAXW 2048
__global__ __launch_bounds__(512)
void wq_merge_kernel(const float* __restrict__ ws,
                     const float* __restrict__ input_mu,
                     float* __restrict__ out, int W)
{
    __shared__ float red[512];
    __shared__ float ew_s[MAXW];
    const int t = threadIdx.x;    // == latent index d

    // global max of per-wave maxima
    float mm = -INFINITY;
    for (int w = t; w < W; w += 512) mm = fmaxf(mm, ws[(size_t)w * WSTRIDE]);
    red[t] = mm; __syncthreads();
    for (int o = 256; o > 0; o >>= 1) {
        if (t < o) red[t] = fmaxf(red[t], red[t + o]);
        __syncthreads();
    }
    const float M = red[0];
    __syncthreads();

    // S = sum_w s_w * exp(m_w - M); stage exp weights in LDS
    float sp = 0.f;
    for (int w = t; w < W; w += 512) {
        float e = __expf(ws[(size_t)w * WSTRIDE] - M);
        ew_s[w] = e;
        sp += ws[(size_t)w * WSTRIDE + 1] * e;
    }
    red[t] = sp; __syncthreads();
    for (int o = 256; o > 0; o >>= 1) {
        if (t < o) red[t] += red[t + o];
        __syncthreads();
    }
    const float S = red[0];
    __syncthreads();

    // quantised[d] = (sum_w ew_w * q_w[d]) / S
    float Qd = 0.f;
    for (int w = 0; w < W; ++w)
        Qd = fmaf(ew_s[w], ws[(size_t)w * WSTRIDE + 2 + t], Qd);
    const float quant = Qd / S;
    out[t] = quant;

    // losses: forward value of both is mean((quantised - input_mu)^2)
    float diff = quant - input_mu[t];
    red[t] = diff * diff; __syncthreads();
    for (int o = 256; o > 0; o >>= 1) {
        if (t < o) red[t] += red[t + o];
        __syncthreads();
    }
    if (t == 0) {
        float l = red[0] * (1.0f / LATENT);
        out[LATENT]     = l;   // loss_enc
        out[LATENT + 1] = l;   // loss_ref
    }
}

// ---------------------------------------------------------------------------
extern "C" void kernel_launch(void* const* d_in, const int* in_sizes, int n_in,
                              void* d_out, int out_size, void* d_ws, size_t ws_size,
                              hipStream_t stream)
{
    (void)in_sizes; (void)n_in; (void)out_size;
    const float* input_mu  = (const float*)d_in[0];
    const float* input_sig = (const float*)d_in[1];
    const float* on_states = (const float*)d_in[2];
    float* out = (float*)d_out;
    float* ws  = (float*)d_ws;

    // pick largest power-of-two wave count that fits the workspace
    int W = MAXW;
    while (W > 1 && (size_t)W * WSTRIDE * sizeof(float) > ws_size) W >>= 1;
    const int tiles_per_wave = N_TILES / W;   // W | 4096 (both powers of two)

    wq_main_kernel<<<W, 32, 0, stream>>>(on_states, input_mu, input_sig,
                                         out + LATENT + 2, ws, tiles_per_wave);
    wq_merge_kernel<<<1, 512, 0, stream>>>(ws, input_mu, out, W);
}